// Crop_21345987461560
// MI455X (gfx1250) — compile-verified
//
#include <hip/hip_runtime.h>

// Crop: out[i,j] = audio[i, j + (j >= starts[i] ? CROP_NUM : 0)]
// B=128, L=262144, CROP_NUM=26214, OUT_LEN=235930
// Pure HBM-streaming gather: per row = two contiguous segment copies.
// Memory-bound: ~242 MB total traffic -> ~10.4 us at 23.3 TB/s peak.

#define B_ROWS   128
#define L_IN     262144
#define CROP_NUM 26214
#define OUT_LEN  (L_IN - CROP_NUM)   // 235930

typedef float f4  __attribute__((ext_vector_type(4)));              // 16B aligned
typedef float f4u __attribute__((ext_vector_type(4), aligned(4)));  // 4B aligned (src may be misaligned to 16)

#define VPT 4                      // float4 vectors per thread
#define TPB 256                    // 8 waves (wave32)
#define ELEMS_PER_BLOCK (TPB * VPT * 4)   // 4096 floats per block
#define PREFETCH_DIST 8192         // floats ahead (~32 KB)

__global__ __launch_bounds__(TPB) void crop_copy_kernel(
    const float* __restrict__ audio,
    const int*   __restrict__ starts,
    float*       __restrict__ out)
{
    const int row = blockIdx.y;      // 0..B-1
    const int seg = blockIdx.z;      // 0: before crop point, 1: after

    // Block-uniform: compiler keeps this on the SALU (s_load_b32).
    const int s = starts[row];

    const int seg_len = seg ? (OUT_LEN - s) : s;
    if (seg_len <= 0) return;

    const long long dst0 = (long long)row * OUT_LEN + (seg ? s : 0);
    const long long src0 = (long long)row * L_IN    + (seg ? (s + CROP_NUM) : 0);

    float*       __restrict__ dst = out   + dst0;
    const float* __restrict__ src = audio + src0;

    // Align the STORE stream to 16B (hipMalloc base is 256B aligned, so
    // element-offset (dst0 & 3) determines alignment).
    int head = (int)((4 - (dst0 & 3)) & 3);
    if (head > seg_len) head = seg_len;
    const int nvec = (seg_len - head) >> 2;               // full float4s
    const int tail = seg_len - head - (nvec << 2);        // 0..3

    // Vectorized body: NT b128 load (possibly 4B-misaligned src) ->
    // NT b128 store (16B-aligned dst). Touch-once data: keep it out of L2.
    int v = blockIdx.x * (TPB * VPT) + threadIdx.x;
    #pragma unroll
    for (int k = 0; k < VPT; ++k, v += TPB) {
        if (v < nvec) {
            const float* sp = src + head + 4 * v;
            // gfx1250 global_prefetch_b8: speculative, pulls the read stream ahead.
            __builtin_prefetch(sp + PREFETCH_DIST, 0, 1);
            f4 val = __builtin_nontemporal_load((const f4u*)sp);
            __builtin_nontemporal_store(val, (f4*)(dst + head + 4 * v));
        }
    }

    // Head (<=3) and tail (<=3) scalars: one block per segment handles them.
    if (blockIdx.x == 0) {
        const int t = threadIdx.x;
        if (t < head) {
            dst[t] = src[t];
        }
        if (t >= 32 && t < 32 + tail) {              // separate wave from head lanes
            const int e = head + (nvec << 2) + (t - 32);
            dst[e] = src[e];
        }
    }
}

extern "C" void kernel_launch(void* const* d_in, const int* in_sizes, int n_in,
                              void* d_out, int out_size, void* d_ws, size_t ws_size,
                              hipStream_t stream) {
    (void)in_sizes; (void)n_in; (void)out_size; (void)d_ws; (void)ws_size;

    const float* audio  = (const float*)d_in[0];   // [B, L] float32
    const int*   starts = (const int*)  d_in[1];   // [B]    int32
    float*       out    = (float*)      d_out;     // [B, OUT_LEN] float32

    dim3 block(TPB);
    dim3 grid((OUT_LEN + ELEMS_PER_BLOCK - 1) / ELEMS_PER_BLOCK,  // 58 tiles
              B_ROWS,                                             // 128 rows
              2);                                                 // 2 segments/row
    crop_copy_kernel<<<grid, block, 0, stream>>>(audio, starts, out);
}